// SpikeFP64SqrtGuess_43860206027322
// MI455X (gfx1250) — compile-verified
//
#include <hip/hip_runtime.h>
#include <stdint.h>

// SpikeFP64SqrtGuess: rows of 64 {0.0,1.0} floats = FP64 bit patterns.
// Output depends only on exponent bits x[1..11]:
//   e_real = (e_x - 1023) mod 2^11 ; is_odd = e_real&1
//   e_new  = (asr1_11bit(e_real) + 1023) mod 2^11
//   out    = [0, bits(e_new) BE, is_odd * bits(0x6A09E667F3BCD)]
// Memory-bound: ~128MB read + 256MB write => ~16us floor at 23.3 TB/s.
// Zero matrix FLOPs => no WMMA role. CDNA5 paths used:
//  - Tensor Data Mover: ONE tensor_load_to_lds per block gathers the strided
//    2D tile (64 rows x 48B, pitch 256B) into LDS, tracked by TENSORcnt.
//  - Non-temporal coalesced global_store_b128 for the write-once output
//    stream (don't thrash the 192MB L2 with 256MB of streaming writes).

typedef uint32_t v4u __attribute__((ext_vector_type(4)));
typedef uint32_t v8u __attribute__((ext_vector_type(8)));
typedef float    v4f __attribute__((ext_vector_type(4)));   // native vector for NT store

#define ROWS_PER_BLOCK 64
#define THREADS 256
#define STAGE_BYTES_PER_ROW 48                 // 12 elements x 4B (covers 0..11)
#define WORD_TABLE_OFF (ROWS_PER_BLOCK * STAGE_BYTES_PER_ROW)

__global__ __launch_bounds__(THREADS)
void sqrt_guess_expand_kernel(const float* __restrict__ x,
                              float* __restrict__ out,
                              int n_rows)
{
    __shared__ uint8_t smem[WORD_TABLE_OFF + ROWS_PER_BLOCK * 8];

    const int tid  = threadIdx.x;
    const int row0 = blockIdx.x * ROWS_PER_BLOCK;

    // ---- Phase 1: one TDM op stages the 64x12-element tile into LDS.
    //      TDM ignores EXEC, so gate to wave 0 => exactly one DMA per block.
    if (tid < 32) {
        const uint64_t gaddr    = (uint64_t)(uintptr_t)(x + (size_t)row0 * 64);
        const uint32_t lds_addr = (uint32_t)(uintptr_t)&smem[0];
        const uint32_t td1      = (uint32_t)(n_rows - row0); // rows left (OOB->0)

        // D# group 0: count=1 | lds_addr | global_addr[56:0] | type=2
        v4u g0;
        g0[0] = 1u;                                   // count=1, user descriptor
        g0[1] = lds_addr;                             // LDS byte address
        g0[2] = (uint32_t)gaddr;                      // global_addr[31:0]
        g0[3] = ((uint32_t)(gaddr >> 32) & 0x01FFFFFFu) | (2u << 30); // [56:32]|type

        // D# group 1:
        //  w0: wg_mask=0 | data_size=2 (4B) | no barrier/iterate/pad
        //  tensor_dim0=64 @ bits 79:48, tensor_dim1=td1 @ 111:80,
        //  tile_dim0=12 @ 127:112, tile_dim1=64 @ 143:128, tile_dim2=0,
        //  tensor_dim0_stride=64 @ 207:160, tensor_dim1_stride=0 (2D)
        v8u g1;
        g1[0] = (2u << 16);                           // data_size = 4 bytes
        g1[1] = (64u & 0xFFFFu) << 16;                // tensor_dim0 lo16
        g1[2] = (td1 & 0xFFFFu) << 16;                // tensor_dim0 hi16 (=0) | td1 lo16
        g1[3] = (td1 >> 16) | (12u << 16);            // td1 hi16 | tile_dim0=12
        g1[4] = 64u;                                  // tile_dim1=64 | tile_dim2=0
        g1[5] = 64u;                                  // tensor_dim0_stride lo32 = 64
        g1[6] = 0u;                                   // stride0 hi16 | stride1 lo16
        g1[7] = 0u;                                   // stride1 hi32
        asm volatile("tensor_load_to_lds %0, %1"
                     :
                     : "s"(g0), "s"(g1)
                     : "memory");
        asm volatile("s_wait_tensorcnt 0x0" ::: "memory");
    }
    __syncthreads();

    // ---- Phase 2: one lane per row folds 11 exponent bits -> 64-bit out word
    if (tid < ROWS_PER_BLOCK && (row0 + tid) < n_rows) {
        const float* rb = (const float*)&smem[tid * STAGE_BYTES_PER_ROW];
        uint32_t e_x = 0u;
#pragma unroll
        for (int i = 1; i <= 11; ++i)
            e_x |= (rb[i] != 0.0f ? 1u : 0u) << (11 - i);
        const uint32_t e_real = (e_x - 1023u) & 0x7FFu;          // mod 2^11
        const uint32_t is_odd = e_real & 1u;
        const int32_t  se     = ((int32_t)(e_real << 21)) >> 21; // sext 11b
        const uint32_t e_new  = (uint32_t)((se >> 1) + 1023) & 0x7FFu;
        const uint64_t mant   = 0x6A09E667F3BCDull & (0ull - (uint64_t)is_odd);
        const uint64_t word   = ((uint64_t)e_new << 52) | mant;  // bit63 = 0
        *(uint64_t*)&smem[WORD_TABLE_OFF + tid * 8] = word;
    }
    __syncthreads();

    // ---- Phase 3: expand 64 rows x 64 bits -> floats.
    //      float4 index f = j*256 + tid => fully coalesced, non-temporal
    //      global_store_b128 (write-once stream, keep L2 for the read side).
    float* obase = out + (size_t)row0 * 64;
#pragma unroll
    for (int j = 0; j < 4; ++j) {
        const int f = j * THREADS + tid;       // float4 index within tile
        const int r = f >> 4;                  // row within tile
        const int p = (f & 15) * 4;            // first element index in row
        if ((row0 + r) < n_rows) {
            const uint64_t w =
                *(const uint64_t*)&smem[WORD_TABLE_OFF + r * 8];
            // out[p+k] = bit (63-(p+k)) of w
            const uint32_t nib = (uint32_t)(w >> (60 - p)) & 0xFu;
            v4f v;
            v.x = (float)((nib >> 3) & 1u);
            v.y = (float)((nib >> 2) & 1u);
            v.z = (float)((nib >> 1) & 1u);
            v.w = (float)(nib & 1u);
            __builtin_nontemporal_store(v, (v4f*)(obase + (size_t)f * 4));
        }
    }
}

extern "C" void kernel_launch(void* const* d_in, const int* in_sizes, int n_in,
                              void* d_out, int out_size, void* d_ws, size_t ws_size,
                              hipStream_t stream)
{
    const float* x   = (const float*)d_in[0];
    float* out       = (float*)d_out;
    const int n_rows = in_sizes[0] / 64;
    const int blocks = (n_rows + ROWS_PER_BLOCK - 1) / ROWS_PER_BLOCK;
    if (blocks > 0) {
        sqrt_guess_expand_kernel<<<blocks, THREADS, 0, stream>>>(x, out, n_rows);
    }
}